// DTSHLoss_38843684225545
// MI455X (gfx1250) — compile-verified
//
#include <hip/hip_runtime.h>
#include <hip/hip_bf16.h>

#define NN 512
#define BIT 64
#define NCLS 100
#define ALPHA 5.0f
#define LAM 1.0f

typedef float v2f __attribute__((ext_vector_type(2)));
typedef float v8f __attribute__((ext_vector_type(8)));

// ---------------------------------------------------------------------------
// Kernel 1: ip = u @ u^T via V_WMMA_F32_16X16X4_F32.
// One wave per 16x16 tile; 8 waves per 256-thread block; 1024 tiles total.
// A fragment (16x4 f32, 2 VGPRs): lanes 0-15 hold M=lane, K={k0,k0+1};
// lanes 16-31 hold M=lane-16, K={k0+2,k0+3}. B (4x16) mirrors this with the
// column tile's rows of u (B = u^T), so both fragments load the same pattern.
// ---------------------------------------------------------------------------
__global__ __launch_bounds__(256) void gram_wmma_kernel(
    const float* __restrict__ u, float* __restrict__ ip) {
  const int lane = threadIdx.x & 31;
  const int wave = threadIdx.x >> 5;
  const int tile = (blockIdx.x << 3) + wave;      // 0..1023
  const int tm = tile >> 5;                       // 0..31
  const int tn = tile & 31;                       // 0..31
  const int l15 = lane & 15;
  const int khalf = (lane >> 4) << 1;             // 0 or 2
  const int mrow = (tm << 4) + l15;
  const int nrow = (tn << 4) + l15;

  v8f c = {};
#pragma unroll
  for (int k0 = 0; k0 < BIT; k0 += 4) {
    v2f a, b;
    a.x = u[mrow * BIT + k0 + khalf];
    a.y = u[mrow * BIT + k0 + khalf + 1];
    b.x = u[nrow * BIT + k0 + khalf];
    b.y = u[nrow * BIT + k0 + khalf + 1];
    // (neg_a, A, neg_b, B, c_mod, C, reuse_a, reuse_b)
    c = __builtin_amdgcn_wmma_f32_16x16x4_f32(false, a, false, b,
                                              (short)0, c, false, false);
  }

  // C/D layout: VGPR j = rows {tm*16+j (lanes 0-15), tm*16+8+j (lanes 16-31)},
  // N = lane&15 within the tile's 16 columns.
  const int col = (tn << 4) + l15;
  const int rbase = (tm << 4) + ((lane >> 4) << 3);
#pragma unroll
  for (int j = 0; j < 8; ++j) {
    ip[(rbase + j) * NN + col] = c[j];
  }
}

// ---------------------------------------------------------------------------
// Kernel 2: labels[r] = argmax(y[r, :]) (y is one-hot fp32).
// ---------------------------------------------------------------------------
__global__ void label_kernel(const float* __restrict__ y, int* __restrict__ lbl) {
  const int r = blockIdx.x * blockDim.x + threadIdx.x;
  if (r >= NN) return;
  int l = 0;
  for (int c = 0; c < NCLS; ++c) {
    if (y[r * NCLS + c] > 0.5f) l = c;
  }
  lbl[r] = l;
}

// ---------------------------------------------------------------------------
// Kernel 3: one block per row r. ip row + labels in LDS. For each
// (p in pos(r), n in neg(r)): t = clamp(ip[p]-ip[n]-alpha, -100, 50),
// f = log1p(exp(t)) - t, accumulate. Deterministic LDS tree reduction.
// ---------------------------------------------------------------------------
__global__ __launch_bounds__(256) void row_loss_kernel(
    const float* __restrict__ ip, const int* __restrict__ lbl,
    float* __restrict__ row_loss, float* __restrict__ valid) {
  const int r = blockIdx.x;
  const int tid = threadIdx.x;
  __shared__ float sIp[NN];
  __shared__ int sL[NN];
  __shared__ float red[256];

  for (int i = tid; i < NN; i += 256) {
    sIp[i] = ip[r * NN + i];
    sL[i] = lbl[i];
  }
  __syncthreads();

  const int myL = sL[r];
  int npos = 0;
  float acc = 0.0f;
  for (int p = 0; p < NN; ++p) {
    if (sL[p] == myL) {
      ++npos;  // uniform across the block (same count in every thread)
      const float ipp = sIp[p];
#pragma unroll
      for (int nb = 0; nb < 2; ++nb) {
        const int n = tid + nb * 256;
        if (sL[n] != myL) {
          float t = ipp - sIp[n] - ALPHA;
          t = fminf(fmaxf(t, -100.0f), 50.0f);
          acc += __logf(1.0f + __expf(t)) - t;
        }
      }
    }
  }

  red[tid] = acc;
  __syncthreads();
  for (int s = 128; s > 0; s >>= 1) {
    if (tid < s) red[tid] += red[tid + s];
    __syncthreads();
  }
  if (tid == 0) {
    const int nneg = NN - npos;
    const float pc = fmaxf((float)npos * (float)nneg, 1.0f);
    row_loss[r] = red[0] / pc;
    valid[r] = (npos > 0 && nneg > 0) ? 1.0f : 0.0f;
  }
}

// ---------------------------------------------------------------------------
// Kernel 4: quantization penalty partials: sum (u - sign(u))^2 per block.
// 128 blocks x 256 threads = 32768 = N*BIT elements.
// ---------------------------------------------------------------------------
__global__ __launch_bounds__(256) void quant_kernel(
    const float* __restrict__ u, float* __restrict__ qpart) {
  __shared__ float red[256];
  const int tid = threadIdx.x;
  const int i = blockIdx.x * 256 + tid;
  const float v = u[i];
  const float sg = (v > 0.0f) ? 1.0f : ((v < 0.0f) ? -1.0f : 0.0f);
  const float d = v - sg;
  red[tid] = d * d;
  __syncthreads();
  for (int s = 128; s > 0; s >>= 1) {
    if (tid < s) red[tid] += red[tid + s];
    __syncthreads();
  }
  if (tid == 0) qpart[blockIdx.x] = red[0];
}

// ---------------------------------------------------------------------------
// Kernel 5: single-block finalize (deterministic).
// loss1 = count>0 ? sum(valid*row_loss)/max(count,1) : 0
// loss2 = LAM * sum(qpart) / (N*BIT)
// ---------------------------------------------------------------------------
__global__ __launch_bounds__(512) void finalize_kernel(
    const float* __restrict__ row_loss, const float* __restrict__ valid,
    const float* __restrict__ qpart, float* __restrict__ out) {
  __shared__ float sA[NN];
  __shared__ float sB[NN];
  __shared__ float sQ[128];
  const int tid = threadIdx.x;

  sA[tid] = valid[tid] * row_loss[tid];
  sB[tid] = valid[tid];
  if (tid < 128) sQ[tid] = qpart[tid];
  __syncthreads();

  for (int s = 256; s > 0; s >>= 1) {
    if (tid < s) { sA[tid] += sA[tid + s]; sB[tid] += sB[tid + s]; }
    __syncthreads();
  }
  for (int s = 64; s > 0; s >>= 1) {
    if (tid < s) sQ[tid] += sQ[tid + s];
    __syncthreads();
  }
  if (tid == 0) {
    const float count = sB[0];
    const float loss1 = (count > 0.0f) ? (sA[0] / fmaxf(count, 1.0f)) : 0.0f;
    const float loss2 = LAM * sQ[0] / (float)(NN * BIT);
    out[0] = loss1 + loss2;
  }
}

// ---------------------------------------------------------------------------
extern "C" void kernel_launch(void* const* d_in, const int* in_sizes, int n_in,
                              void* d_out, int out_size, void* d_ws, size_t ws_size,
                              hipStream_t stream) {
  (void)in_sizes; (void)n_in; (void)out_size; (void)ws_size;
  const float* u = (const float*)d_in[0];   // [512, 64] fp32
  const float* y = (const float*)d_in[1];   // [512, 100] fp32 one-hot
  float* out = (float*)d_out;               // scalar

  // Workspace layout (bytes):
  char* ws = (char*)d_ws;
  float* ip       = (float*)(ws);                         // 512*512*4 = 1048576
  int*   lbl      = (int*)  (ws + 1048576);               // 512*4
  float* row_loss = (float*)(ws + 1048576 + 2048);        // 512*4
  float* valid    = (float*)(ws + 1048576 + 4096);        // 512*4
  float* qpart    = (float*)(ws + 1048576 + 6144);        // 128*4

  // 1. Gram matrix via WMMA: 1024 tiles, 8 waves/block -> 128 blocks.
  gram_wmma_kernel<<<128, 256, 0, stream>>>(u, ip);
  // 2. Labels from one-hot y.
  label_kernel<<<2, 256, 0, stream>>>(y, lbl);
  // 3. Per-row fused triplet softplus loss.
  row_loss_kernel<<<NN, 256, 0, stream>>>(ip, lbl, row_loss, valid);
  // 4. Quantization penalty partials.
  quant_kernel<<<128, 256, 0, stream>>>(u, qpart);
  // 5. Deterministic finalize.
  finalize_kernel<<<1, 512, 0, stream>>>(row_loss, valid, qpart, out);
}